// CapsuleLayer_29205777613283
// MI455X (gfx1250) — compile-verified
//
#include <hip/hip_runtime.h>
#include <stdint.h>

#define NB 4096
#define SEQ 200
#define SP 208          // SEQ padded to 13 tiles of 16
#define LOWD 64
#define HIGHD 64
#define KCAP 8
#define KP 16
#define ITERS 3
#define XST 68          // row stride (floats) for Xs/Ls/Ws/Hs: 272B rows, 16B aligned, bank-spread
#define RST 17          // row stride for rl (17 coprime with 64 banks)

typedef __attribute__((ext_vector_type(2))) float v2f;
typedef __attribute__((ext_vector_type(8))) float v8f;
typedef __attribute__((ext_vector_type(4))) int   v4i;
typedef v4i __attribute__((address_space(1)))* gv4i_p;   // global int4*
typedef v4i __attribute__((address_space(3)))* lv4i_p;   // LDS int4*

#if defined(__has_builtin)
#if __has_builtin(__builtin_amdgcn_global_load_async_to_lds_b128)
#define HAVE_ASYNC_LDS 1
#endif
#endif

__device__ __forceinline__ uint32_t hash_u32(uint32_t x) {
    x ^= x >> 16; x *= 0x7feb352du;
    x ^= x >> 15; x *= 0x846ca68bu;
    x ^= x >> 16;
    return x;
}

// deterministic standard normal per index (Box-Muller)
__device__ __forceinline__ float gauss(uint32_t seed) {
    uint32_t h1 = hash_u32(seed ^ 0x2545F491u);
    uint32_t h2 = hash_u32(seed + 0x9E3779B9u);
    float u1 = (float)(h1 >> 8) * (1.0f / 16777216.0f) + 1e-7f;
    float u2 = (float)(h2 >> 8) * (1.0f / 16777216.0f);
    float r = sqrtf(-2.0f * __logf(u1));
    return r * __cosf(6.28318530718f * u2);
}

__device__ __forceinline__ float wave_max(float v) {
    #pragma unroll
    for (int o = 16; o > 0; o >>= 1) v = fmaxf(v, __shfl_xor(v, o, 32));
    return v;
}
__device__ __forceinline__ float wave_sum(float v) {
    #pragma unroll
    for (int o = 16; o > 0; o >>= 1) v += __shfl_xor(v, o, 32);
    return v;
}

__device__ __forceinline__ v8f wmma_f32(v2f a, v2f b, v8f c) {
    return __builtin_amdgcn_wmma_f32_16x16x4_f32(false, a, false, b, (short)0, c,
                                                 false, false);
}

__global__ __launch_bounds__(256) void capsule_route_kernel(
    const float* __restrict__ inputs,   // [NB, SEQ, LOWD]
    const int*   __restrict__ seq_len,  // [NB]
    const float* __restrict__ Wmat,     // [LOWD, HIGHD]
    float*       __restrict__ out)      // [NB, KCAP, HIGHD]
{
    __shared__ float Xs[SP * XST];      // staged inputs[b]
    __shared__ float Ls[SP * XST];      // low = X @ W
    __shared__ float Ws[LOWD * XST];    // staged bilinear matrix
    __shared__ float Rl[SP * RST];      // routing logits (cols 8..15 zero)
    __shared__ float Hs[KP * XST];      // high capsules (rows 8..15 zero)
    __shared__ float Sc[KCAP];          // squash scales

    const int b    = blockIdx.x;
    const int tid  = threadIdx.x;
    const int wave = tid >> 5;
    const int lane = tid & 31;
    const int lrow = lane & 15;
    const int lhi  = lane >> 4;

    const int   slen  = seq_len[b];
    const float nhigh = fminf(fmaxf(log2f((float)slen), 1.0f), 8.0f);

    // ---------------- stage X_b and W into LDS (async tensor path) ----------
    const float* inp_b = inputs + (size_t)b * (SEQ * LOWD);
    for (int idx = tid; idx < (SEQ * LOWD) / 4; idx += 256) {   // 3200 x float4
        int row = idx >> 4;
        int col = (idx & 15) << 2;
        int dst = row * XST + col;
#if HAVE_ASYNC_LDS
        __builtin_amdgcn_global_load_async_to_lds_b128(
            (gv4i_p)(inp_b + (idx << 2)), (lv4i_p)&Xs[dst], 0, 0);
#else
        *(float4*)&Xs[dst] = *(const float4*)(inp_b + (idx << 2));
#endif
    }
    for (int idx = tid; idx < (LOWD * HIGHD) / 4; idx += 256) { // 1024 x float4
        int row = idx >> 4;
        int col = (idx & 15) << 2;
        int dst = row * XST + col;
#if HAVE_ASYNC_LDS
        __builtin_amdgcn_global_load_async_to_lds_b128(
            (gv4i_p)(Wmat + (idx << 2)), (lv4i_p)&Ws[dst], 0, 0);
#else
        *(float4*)&Ws[dst] = *(const float4*)(Wmat + (idx << 2));
#endif
    }
    // zero the padded tail rows of Xs (s = 200..207)
    for (int idx = tid; idx < 8 * XST; idx += 256) Xs[SEQ * XST + idx] = 0.0f;
    // init routing logits: N(0,1) for s<200, k<8; zero padding elsewhere
    for (int idx = tid; idx < SP * RST; idx += 256) {
        int s = idx / RST, k = idx % RST;
        float v = 0.0f;
        if (s < SEQ && k < KCAP)
            v = gauss(((uint32_t)b * SEQ + (uint32_t)s) * KCAP + (uint32_t)k);
        Rl[idx] = v;
    }
#if HAVE_ASYNC_LDS
#if __has_builtin(__builtin_amdgcn_s_wait_asynccnt)
    __builtin_amdgcn_s_wait_asynccnt(0);
#else
    asm volatile("s_wait_asynccnt 0" ::: "memory");
#endif
#endif
    __syncthreads();

    // ---------------- GEMM1: Ls[SPxHIGH] = Xs[SPxLOW] @ Ws[LOWxHIGH] --------
    for (int t = wave; t < 13 * 4; t += 8) {
        const int m0 = (t >> 2) << 4;
        const int n0 = (t & 3) << 4;
        v8f acc = {};
        #pragma unroll
        for (int k0 = 0; k0 < LOWD; k0 += 4) {
            v2f a, bb;
            const int ar = (m0 + lrow) * XST + k0 + 2 * lhi;
            a.x = Xs[ar];
            a.y = Xs[ar + 1];
            const int br = (k0 + 2 * lhi) * XST + n0 + lrow;
            bb.x = Ws[br];
            bb.y = Ws[br + XST];
            acc = wmma_f32(a, bb, acc);
        }
        #pragma unroll
        for (int r = 0; r < 8; ++r)
            Ls[(m0 + r + 8 * lhi) * XST + n0 + lrow] = acc[r];
    }
    __syncthreads();

    // ---------------- dynamic routing ---------------------------------------
    for (int iter = 0; iter < ITERS; ++iter) {
        // (a) clamp + softmax over sequence + seq mask; one wave per capsule
        {
            const int k = wave;
            const float thr = ((float)k < nhigh) ? 1e32f : -1e32f;
            float mx = -3.4e38f;
            for (int s = lane; s < SEQ; s += 32) {
                float v = fminf(Rl[s * RST + k], thr);
                Rl[s * RST + k] = v;
                mx = fmaxf(mx, v);
            }
            mx = wave_max(mx);
            float sum = 0.0f;
            for (int s = lane; s < SEQ; s += 32) {
                float e = __expf(Rl[s * RST + k] - mx);
                Rl[s * RST + k] = e;
                sum += e;
            }
            sum = wave_sum(sum);
            const float inv = 1.0f / sum;
            for (int s = lane; s < SEQ; s += 32) {
                float w = Rl[s * RST + k] * inv;
                Rl[s * RST + k] = (s < slen) ? w : 0.0f;
            }
        }
        __syncthreads();

        // (b) Hs[KPxHIGH] = Rl^T[KPxSP] @ Ls[SPxHIGH]; waves 0..3 (one N-tile each)
        if (wave < 4) {
            const int n0 = wave << 4;
            v8f acc = {};
            for (int k0 = 0; k0 < SP; k0 += 4) {
                v2f a, bb;
                a.x = Rl[(k0 + 2 * lhi) * RST + lrow];       // A[m][kk]=rl[s][m]
                a.y = Rl[(k0 + 2 * lhi + 1) * RST + lrow];
                const int br = (k0 + 2 * lhi) * XST + n0 + lrow;
                bb.x = Ls[br];
                bb.y = Ls[br + XST];
                acc = wmma_f32(a, bb, acc);
            }
            #pragma unroll
            for (int r = 0; r < 8; ++r)
                Hs[(r + 8 * lhi) * XST + n0 + lrow] = acc[r];
        }
        __syncthreads();

        // (c) squash: scale_k = n/(1+n^2), n = max(||high_k||, 1e-7)
        if (tid < KCAP) {
            float ss = 0.0f;
            #pragma unroll 8
            for (int h = 0; h < HIGHD; ++h) {
                float v = Hs[tid * XST + h];
                ss += v * v;
            }
            float n = fmaxf(sqrtf(ss), 1e-7f);
            Sc[tid] = n / (1.0f + n * n);
        }
        __syncthreads();
        for (int idx = tid; idx < KCAP * HIGHD; idx += 256) {
            int k = idx >> 6, h = idx & 63;
            Hs[k * XST + h] *= Sc[k];
        }
        __syncthreads();

        // (d) Rl[SPxKP] += Ls[SPxHIGH] @ Hs^T[HIGHxKP]  (dead on last iter)
        if (iter < ITERS - 1) {
            for (int mt = wave; mt < 13; mt += 8) {
                const int m0 = mt << 4;
                v8f acc;
                #pragma unroll
                for (int r = 0; r < 8; ++r)
                    acc[r] = Rl[(m0 + r + 8 * lhi) * RST + lrow];   // C = rl
                #pragma unroll
                for (int k0 = 0; k0 < HIGHD; k0 += 4) {
                    v2f a, bb;
                    const int ar = (m0 + lrow) * XST + k0 + 2 * lhi;
                    a.x = Ls[ar];
                    a.y = Ls[ar + 1];
                    const int br = lrow * XST + k0 + 2 * lhi;       // B[kk][n]=high[n][kk]
                    bb.x = Hs[br];
                    bb.y = Hs[br + 1];
                    acc = wmma_f32(a, bb, acc);
                }
                #pragma unroll
                for (int r = 0; r < 8; ++r)
                    Rl[(m0 + r + 8 * lhi) * RST + lrow] = acc[r];
            }
            __syncthreads();
        }
    }

    // ---------------- write high[b] -----------------------------------------
    for (int idx = tid; idx < KCAP * HIGHD; idx += 256) {
        int k = idx >> 6, h = idx & 63;
        out[(size_t)b * (KCAP * HIGHD) + idx] = Hs[k * XST + h];
    }
}

extern "C" void kernel_launch(void* const* d_in, const int* in_sizes, int n_in,
                              void* d_out, int out_size, void* d_ws, size_t ws_size,
                              hipStream_t stream) {
    const float* inputs  = (const float*)d_in[0];
    const int*   seq_len = (const int*)d_in[1];
    const float* Wmat    = (const float*)d_in[2];
    float*       out     = (float*)d_out;
    hipLaunchKernelGGL(capsule_route_kernel, dim3(NB), dim3(256), 0, stream,
                       inputs, seq_len, Wmat, out);
}